// _VisionMambaDualHead_52802327937057
// MI455X (gfx1250) — compile-verified
//
#include <hip/hip_runtime.h>
#include <hip/hip_bf16.h>
#include <math.h>

// ---------------------------------------------------------------------------
// VisionMamba dual-head forward for gfx1250 (MI455X).
// fp32 throughout (matches reference); GEMMs via V_WMMA_F32_16X16X4_F32,
// 16x64 output strip per wave, K unrolled by 16 (16 WMMAs per address calc).
// ---------------------------------------------------------------------------

#define BATCH 32
#define DMODEL 192
#define DEPTH 4
#define DINNER 384
#define DSTATE 16
#define DCONV 4
#define PATCH 16
#define IMG 224
#define NPATCH 196          // (224/16)^2
#define LSEQ 197            // N + 1 (cls token)
#define NCLS 2
#define DHALF 96            // D/2 for reg head
#define KPATCH 768          // 3*16*16

typedef __attribute__((ext_vector_type(2))) float v2f;
typedef __attribute__((ext_vector_type(8))) float v8f;

// ---------------------------------------------------------------------------
// Patch gather: x (B,3,224,224) -> patches (B*196, 768), j = c*256+py*16+px
// ---------------------------------------------------------------------------
__global__ void k_gather_patches(const float* __restrict__ x,
                                 float* __restrict__ patches) {
  int idx = blockIdx.x * blockDim.x + threadIdx.x;
  const int total = BATCH * NPATCH * KPATCH;
  if (idx >= total) return;
  int j    = idx % KPATCH;
  int rest = idx / KPATCH;
  int n    = rest % NPATCH;
  int b    = rest / NPATCH;
  int c  = j / (PATCH * PATCH);
  int pj = j % (PATCH * PATCH);
  int py = pj / PATCH, px = pj % PATCH;
  int hy = n / (IMG / PATCH), hx = n % (IMG / PATCH);
  int row = hy * PATCH + py, col = hx * PATCH + px;
  patches[idx] = x[(((size_t)b * 3 + c) * IMG + row) * IMG + col];
}

// ---------------------------------------------------------------------------
// fp32 WMMA GEMM:  C[M,N] (+)= A[M,K] * W[N,K]^T (+ bias[N])
// One wave computes a 16x64 strip (4 accumulators share each A fragment).
// K stepped by 16: one address calc -> clause of 20 b64 loads -> 16 WMMAs.
// Out-of-range N columns: wrow clamped to a valid row (loads legal); the
// resulting garbage stays confined to its own D column, which is never
// stored (store guard col < N) -- so no per-iteration zeroing needed.
// Lane mapping (V_WMMA_F32_16X16X4_F32):
//   A frag: row m = tile_m*16 + (lane&15); K pair (2*half, 2*half+1)
//   B frag: col n = (lane&15) of W row n; same K pair
//   D frag: col (lane&15), rows v + 8*half, v in [0,8)
// ---------------------------------------------------------------------------
__global__ void k_gemm_wmma_f32(const float* __restrict__ A,
                                const float* __restrict__ W,
                                const float* __restrict__ bias,
                                float* __restrict__ C,
                                int M, int N, int K, int accumulate) {
  const int lane = threadIdx.x;        // 0..31, full wave (wave32)
  const int half = lane >> 4;          // 0 or 1
  const int id16 = lane & 15;
  const int m  = blockIdx.y * 16 + id16;
  const int n0 = blockIdx.x * 64;
  const int mc = (m < M) ? m : (M - 1);

  const float* __restrict__ arow = A + (size_t)mc * K;

  const float* wrow[4];
#pragma unroll
  for (int j = 0; j < 4; ++j) {
    int n = n0 + j * 16 + id16;
    int nc = (n < N) ? n : (N - 1);   // clamp: loads stay in-bounds
    wrow[j] = W + (size_t)nc * K;
  }

  v8f acc[4];
#pragma unroll
  for (int j = 0; j < 4; ++j)
#pragma unroll
    for (int v = 0; v < 8; ++v) acc[j][v] = 0.0f;

  const int kbase = 2 * half;
  for (int k = 0; k < K; k += 16) {   // K is a multiple of 16 for all calls
    if (k + 64 < K)
      __builtin_prefetch(&arow[k + 64], 0, 3);  // WGP-scope prefetch of A
    v2f a[4], b[4][4];
#pragma unroll
    for (int u = 0; u < 4; ++u) {
      const int kk = k + 4 * u + kbase;
      a[u].x = arow[kk];
      a[u].y = arow[kk + 1];
#pragma unroll
      for (int j = 0; j < 4; ++j) {
        b[u][j].x = wrow[j][kk];
        b[u][j].y = wrow[j][kk + 1];
      }
    }
#pragma unroll
    for (int u = 0; u < 4; ++u)
#pragma unroll
      for (int j = 0; j < 4; ++j)
        acc[j] = __builtin_amdgcn_wmma_f32_16x16x4_f32(
            /*neg_a=*/false, a[u], /*neg_b=*/false, b[u][j],
            /*c_mod=*/(short)0, acc[j], /*reuse_a=*/false, /*reuse_b=*/false);
  }

#pragma unroll
  for (int j = 0; j < 4; ++j) {
    int col = n0 + j * 16 + id16;
    float bv = 0.0f;
    if (bias != nullptr && col < N) bv = bias[col];
#pragma unroll
    for (int v = 0; v < 8; ++v) {
      int row = blockIdx.y * 16 + v + 8 * half;
      if (row < M && col < N) {
        size_t off = (size_t)row * N + col;
        float prev = accumulate ? C[off] : 0.0f;
        C[off] = prev + acc[j][v] + bv;
      }
    }
  }
}

// ---------------------------------------------------------------------------
// t[b,l,:] = (l==0 ? cls_token : tok[b,l-1,:]) + pos_embed[l,:]
// ---------------------------------------------------------------------------
__global__ void k_assemble_t(const float* __restrict__ tok,
                             const float* __restrict__ cls_token,
                             const float* __restrict__ pos,
                             float* __restrict__ t) {
  int idx = blockIdx.x * blockDim.x + threadIdx.x;
  const int total = BATCH * LSEQ * DMODEL;
  if (idx >= total) return;
  int d = idx % DMODEL;
  int l = (idx / DMODEL) % LSEQ;
  int b = idx / (DMODEL * LSEQ);
  float v = (l == 0) ? cls_token[d]
                     : tok[((size_t)b * NPATCH + (l - 1)) * DMODEL + d];
  t[idx] = v + pos[l * DMODEL + d];
}

// ---------------------------------------------------------------------------
// LayerNorm per row (dim=192): one wave per row, 6 elems/lane, shfl reduce.
// ---------------------------------------------------------------------------
__global__ void k_layernorm(const float* __restrict__ x,
                            const float* __restrict__ g,
                            const float* __restrict__ bb,
                            float* __restrict__ out) {
  const int row = blockIdx.x;
  const int lane = threadIdx.x;  // 0..31
  const float* __restrict__ xr = x + (size_t)row * DMODEL;
  float vals[6];
  float s = 0.0f;
#pragma unroll
  for (int i = 0; i < 6; ++i) {
    vals[i] = xr[lane + i * 32];
    s += vals[i];
  }
#pragma unroll
  for (int off = 16; off > 0; off >>= 1) s += __shfl_xor(s, off, 32);
  float m = s * (1.0f / DMODEL);
  float q = 0.0f;
#pragma unroll
  for (int i = 0; i < 6; ++i) {
    float d = vals[i] - m;
    q += d * d;
  }
#pragma unroll
  for (int off = 16; off > 0; off >>= 1) q += __shfl_xor(q, off, 32);
  float inv = rsqrtf(q * (1.0f / DMODEL) + 1e-5f);
#pragma unroll
  for (int i = 0; i < 6; ++i) {
    int d = lane + i * 32;
    out[(size_t)row * DMODEL + d] = (vals[i] - m) * inv * g[d] + bb[d];
  }
}

// ---------------------------------------------------------------------------
// Causal depthwise conv (DC=4) over L + bias + SiLU.
// xc = xz[..., :384];   xa[b,l,c] = silu(sum_k cw[c,k]*xc[b,l-3+k,c] + cb[c])
// ---------------------------------------------------------------------------
__global__ void k_conv_silu(const float* __restrict__ xz,
                            const float* __restrict__ cw,
                            const float* __restrict__ cb,
                            float* __restrict__ xa) {
  int idx = blockIdx.x * blockDim.x + threadIdx.x;
  const int total = BATCH * LSEQ * DINNER;
  if (idx >= total) return;
  int c = idx % DINNER;
  int l = (idx / DINNER) % LSEQ;
  int b = idx / (DINNER * LSEQ);
  float acc = cb[c];
#pragma unroll
  for (int k = 0; k < DCONV; ++k) {
    int lp = l - (DCONV - 1) + k;
    if (lp >= 0)
      acc += cw[c * DCONV + k] * xz[((size_t)b * LSEQ + lp) * (2 * DINNER) + c];
  }
  xa[idx] = acc / (1.0f + expf(-acc));  // silu
}

// ---------------------------------------------------------------------------
// Selective-scan: one thread per (b, c); h[16] in registers; 197 serial steps.
// yz[b,l,c] = (sum_n h*C + xa*D) * silu(z)
// ---------------------------------------------------------------------------
__global__ void k_scan(const float* __restrict__ bcd,
                       const float* __restrict__ xa,
                       const float* __restrict__ xz,
                       const float* __restrict__ w_dt,
                       const float* __restrict__ b_dt,
                       const float* __restrict__ A_log,
                       const float* __restrict__ D_ssm,
                       float* __restrict__ yz) {
  int tid = blockIdx.x * blockDim.x + threadIdx.x;
  if (tid >= BATCH * DINNER) return;
  int b = tid / DINNER;
  int c = tid % DINNER;
  float wdt = w_dt[c], bdt = b_dt[c], dssm = D_ssm[c];
  float negA[DSTATE];
#pragma unroll
  for (int n = 0; n < DSTATE; ++n) negA[n] = -expf(A_log[c * DSTATE + n]);
  float h[DSTATE];
#pragma unroll
  for (int n = 0; n < DSTATE; ++n) h[n] = 0.0f;

  for (int l = 0; l < LSEQ; ++l) {
    const size_t rl = (size_t)b * LSEQ + l;
    const float* __restrict__ bl = bcd + rl * (2 * DSTATE + 1);
    float dtraw = bl[2 * DSTATE] * wdt + bdt;
    float dt = (dtraw > 20.0f) ? dtraw : log1pf(expf(dtraw));  // softplus
    float xav = xa[rl * DINNER + c];
    float zv = xz[rl * (2 * DINNER) + DINNER + c];
    float y = 0.0f;
#pragma unroll
    for (int n = 0; n < DSTATE; ++n) {
      float dA = expf(dt * negA[n]);
      h[n] = dA * h[n] + dt * bl[n] * xav;
      y += h[n] * bl[DSTATE + n];
    }
    y += xav * dssm;
    float sz = zv / (1.0f + expf(-zv));  // silu(z)
    yz[rl * DINNER + c] = y * sz;
  }
}

// ---------------------------------------------------------------------------
// Heads: final LN on token 0, cls = feat@cls_w^T+b, reg = gelu(feat@W1^T+b1)@W2^T+b2
// One block of 32 threads, one batch element per thread; feat staged in LDS.
// ---------------------------------------------------------------------------
__global__ void k_heads(const float* __restrict__ t,
                        const float* __restrict__ fn_g,
                        const float* __restrict__ fn_b,
                        const float* __restrict__ cls_w,
                        const float* __restrict__ cls_b,
                        const float* __restrict__ reg_w1,
                        const float* __restrict__ reg_b1,
                        const float* __restrict__ reg_w2,
                        const float* __restrict__ reg_b2,
                        float* __restrict__ out) {
  __shared__ float feat[BATCH][DMODEL + 1];
  const int b = threadIdx.x;  // 0..31
  const float* __restrict__ row = t + (size_t)b * LSEQ * DMODEL;  // token 0
  float m = 0.0f;
  for (int d = 0; d < DMODEL; ++d) m += row[d];
  m *= (1.0f / DMODEL);
  float v = 0.0f;
  for (int d = 0; d < DMODEL; ++d) {
    float q = row[d] - m;
    v += q * q;
  }
  float inv = rsqrtf(v * (1.0f / DMODEL) + 1e-5f);
  for (int d = 0; d < DMODEL; ++d)
    feat[b][d] = (row[d] - m) * inv * fn_g[d] + fn_b[d];

  // cls head
  for (int j = 0; j < NCLS; ++j) {
    float s = cls_b[j];
    for (int d = 0; d < DMODEL; ++d) s += feat[b][d] * cls_w[j * DMODEL + d];
    out[b * NCLS + j] = s;
  }
  // reg head: exact gelu = 0.5*x*(1+erf(x/sqrt(2)))
  float r = reg_b2[0];
  for (int j = 0; j < DHALF; ++j) {
    float hsum = reg_b1[j];
    for (int d = 0; d < DMODEL; ++d) hsum += feat[b][d] * reg_w1[j * DMODEL + d];
    float ge = 0.5f * hsum * (1.0f + erff(hsum * 0.70710678118654752f));
    r += ge * reg_w2[j];
  }
  out[BATCH * NCLS + b] = r;
}

// ---------------------------------------------------------------------------
// Host orchestration
// ---------------------------------------------------------------------------
extern "C" void kernel_launch(void* const* d_in, const int* in_sizes, int n_in,
                              void* d_out, int out_size, void* d_ws, size_t ws_size,
                              hipStream_t stream) {
  const float* x         = (const float*)d_in[0];
  const float* patch_w   = (const float*)d_in[1];
  const float* patch_b   = (const float*)d_in[2];
  const float* cls_token = (const float*)d_in[3];
  const float* pos_embed = (const float*)d_in[4];
  const float* ln_g      = (const float*)d_in[5];
  const float* ln_b      = (const float*)d_in[6];
  const float* w_in      = (const float*)d_in[7];
  const float* conv_w    = (const float*)d_in[8];
  const float* conv_b    = (const float*)d_in[9];
  const float* w_x       = (const float*)d_in[10];
  const float* w_dt      = (const float*)d_in[11];
  const float* b_dt      = (const float*)d_in[12];
  const float* A_log     = (const float*)d_in[13];
  const float* D_ssm     = (const float*)d_in[14];
  const float* w_out     = (const float*)d_in[15];
  const float* fn_g      = (const float*)d_in[16];
  const float* fn_b      = (const float*)d_in[17];
  const float* cls_w     = (const float*)d_in[18];
  const float* cls_b     = (const float*)d_in[19];
  const float* reg_w1    = (const float*)d_in[20];
  const float* reg_b1    = (const float*)d_in[21];
  const float* reg_w2    = (const float*)d_in[22];
  const float* reg_b2    = (const float*)d_in[23];
  float* out = (float*)d_out;

  // Workspace layout (fp32, 256B-aligned slices; total ~74 MB)
  size_t off = 0;
  auto carve = [&](size_t elems) {
    size_t o = off;
    off += ((elems * sizeof(float)) + 255) & ~(size_t)255;
    return (float*)((char*)d_ws + o);
  };
  const int MROWS = BATCH * LSEQ;  // 6304
  float* patches = carve((size_t)BATCH * NPATCH * KPATCH);   // 19.3 MB
  float* tok     = carve((size_t)BATCH * NPATCH * DMODEL);   //  4.8 MB
  float* t       = carve((size_t)MROWS * DMODEL);            //  4.8 MB
  float* u       = carve((size_t)MROWS * DMODEL);            //  4.8 MB
  float* xz      = carve((size_t)MROWS * 2 * DINNER);        // 19.4 MB
  float* xa      = carve((size_t)MROWS * DINNER);            //  9.7 MB
  float* bcd     = carve((size_t)MROWS * (2 * DSTATE + 1));  //  0.8 MB
  float* yz      = carve((size_t)MROWS * DINNER);            //  9.7 MB
  (void)ws_size; (void)n_in; (void)in_sizes; (void)out_size;

  // 1) Patchify + embed
  {
    int total = BATCH * NPATCH * KPATCH;
    k_gather_patches<<<(total + 255) / 256, 256, 0, stream>>>(x, patches);
    dim3 grid((DMODEL + 63) / 64, (BATCH * NPATCH) / 16);  // (3, 392)
    k_gemm_wmma_f32<<<grid, 32, 0, stream>>>(patches, patch_w, patch_b, tok,
                                             BATCH * NPATCH, DMODEL, KPATCH, 0);
    int tt = BATCH * LSEQ * DMODEL;
    k_assemble_t<<<(tt + 255) / 256, 256, 0, stream>>>(tok, cls_token,
                                                       pos_embed, t);
  }

  // 2) Mamba layers
  for (int i = 0; i < DEPTH; ++i) {
    k_layernorm<<<MROWS, 32, 0, stream>>>(t, ln_g + i * DMODEL,
                                          ln_b + i * DMODEL, u);
    {
      dim3 grid((2 * DINNER + 63) / 64, MROWS / 16);  // (12, 394)
      k_gemm_wmma_f32<<<grid, 32, 0, stream>>>(
          u, w_in + (size_t)i * 2 * DINNER * DMODEL, nullptr, xz,
          MROWS, 2 * DINNER, DMODEL, 0);
    }
    {
      int total = MROWS * DINNER;
      k_conv_silu<<<(total + 255) / 256, 256, 0, stream>>>(
          xz, conv_w + (size_t)i * DINNER * DCONV, conv_b + (size_t)i * DINNER,
          xa);
    }
    {
      dim3 grid((2 * DSTATE + 1 + 63) / 64, MROWS / 16);  // (1, 394)
      k_gemm_wmma_f32<<<grid, 32, 0, stream>>>(
          xa, w_x + (size_t)i * (2 * DSTATE + 1) * DINNER, nullptr, bcd,
          MROWS, 2 * DSTATE + 1, DINNER, 0);
    }
    {
      int total = BATCH * DINNER;  // 12288
      k_scan<<<(total + 255) / 256, 256, 0, stream>>>(
          bcd, xa, xz, w_dt + (size_t)i * DINNER, b_dt + (size_t)i * DINNER,
          A_log + (size_t)i * DINNER * DSTATE, D_ssm + (size_t)i * DINNER, yz);
    }
    {
      dim3 grid((DMODEL + 63) / 64, MROWS / 16);  // (3, 394), accumulate resid
      k_gemm_wmma_f32<<<grid, 32, 0, stream>>>(
          yz, w_out + (size_t)i * DMODEL * DINNER, nullptr, t,
          MROWS, DMODEL, DINNER, 1);
    }
  }

  // 3) Final LN (token 0 only) + heads
  k_heads<<<1, 32, 0, stream>>>(t, fn_g, fn_b, cls_w, cls_b, reg_w1, reg_b1,
                                reg_w2, reg_b2, out);
}